// LstmEncoder_88321707475560
// MI455X (gfx1250) — compile-verified
//
#include <hip/hip_runtime.h>
#include <math.h>

typedef __attribute__((ext_vector_type(2))) float v2f;
typedef __attribute__((ext_vector_type(8))) float v8f;

#define B_   32
#define T_   256
#define H_   512
#define G_   2048
#define NWG  16

// ---------------- workspace layout (float offsets) ----------------
static constexpr size_t OFF_E      = 0;                         // 16 floats
static constexpr size_t OFF_BAR    = 32;                        // unsigned[8]
static constexpr size_t OFF_H      = 64;                        // 4 dirs x 2 bufs x 32*512
static constexpr size_t OFF_BIAS   = OFF_H + 4ull*2*B_*H_;      // 4 x 2048
static constexpr size_t OFF_W0F_IH = OFF_BIAS + 4ull*G_;
static constexpr size_t OFF_W0F_HH = OFF_W0F_IH + (size_t)G_*512;
static constexpr size_t OFF_W0B_IH = OFF_W0F_HH + (size_t)G_*H_;
static constexpr size_t OFF_W0B_HH = OFF_W0B_IH + (size_t)G_*512;
static constexpr size_t OFF_W1F_IH = OFF_W0B_HH + (size_t)G_*H_;
static constexpr size_t OFF_W1F_HH = OFF_W1F_IH + (size_t)G_*1024;
static constexpr size_t OFF_W1B_IH = OFF_W1F_HH + (size_t)G_*H_;
static constexpr size_t OFF_W1B_HH = OFF_W1B_IH + (size_t)G_*1024;
static constexpr size_t OFF_X1     = OFF_W1B_HH + (size_t)G_*H_;
// total = OFF_X1 + 32*256*1024 floats ~= 76 MB

// ---------------- embedding renorm ----------------
__global__ void embed_kernel(const float* __restrict__ emb,
                             const int* __restrict__ domain,
                             float* __restrict__ e_out)
{
    if (threadIdx.x == 0) {
        int d = domain[0];
        float v[16];
        float s = 0.f;
        #pragma unroll
        for (int m = 0; m < 16; ++m) { v[m] = emb[d*16 + m]; s += v[m]*v[m]; }
        float n = sqrtf(s);
        float scale = fminf(1.f, 1.f / fmaxf(n, 1e-7f));
        #pragma unroll
        for (int m = 0; m < 16; ++m) e_out[m] = v[m]*scale;
    }
}

// ---------------- weight generation: out[j] = dot16(w[j*16..], e) ----------------
__global__ void gen_weight(const float* __restrict__ w, const float* __restrict__ e,
                           float* __restrict__ out, int n)
{
    int i = blockIdx.x*blockDim.x + threadIdx.x;
    if (i >= n) return;
    const float4* wp = (const float4*)(w + (size_t)i*16);
    const float4* ep = (const float4*)e;
    float s = 0.f;
    #pragma unroll
    for (int q = 0; q < 4; ++q) {
        float4 wv = wp[q], ev = ep[q];
        s += wv.x*ev.x + wv.y*ev.y + wv.z*ev.z + wv.w*ev.w;
    }
    out[i] = s;
}

__global__ void gen_bias(const float* __restrict__ bih, const float* __restrict__ bhh,
                         const float* __restrict__ e, float* __restrict__ out, int n)
{
    int i = blockIdx.x*blockDim.x + threadIdx.x;
    if (i >= n) return;
    const float4* p0 = (const float4*)(bih + (size_t)i*16);
    const float4* p1 = (const float4*)(bhh + (size_t)i*16);
    const float4* ep = (const float4*)e;
    float s = 0.f;
    #pragma unroll
    for (int q = 0; q < 4; ++q) {
        float4 a = p0[q], b = p1[q], ev = ep[q];
        s += (a.x+b.x)*ev.x + (a.y+b.y)*ev.y + (a.z+b.z)*ev.z + (a.w+b.w)*ev.w;
    }
    out[i] = s;
}

// ---------------- fused bidirectional LSTM scan (one layer) ----------------
// grid = 2*NWG blocks: blocks [0,NWG) = forward dir, [NWG,2*NWG) = backward dir.
// Each WG owns 32 hidden units -> 4x32 = 128 gate rows. 16 waves, each computes
// one 16(batch) x 16(gatecol) tile with V_WMMA_F32_16X16X4_F32 over K = DX + 512.
// DX is a compile-time constant (512 or 1024) so both K-loops have static trip
// counts and the step body pipelines without zero-trip guards.
template<int DX>
__global__ void __launch_bounds__(512)
lstm_scan(const float* __restrict__ X,
          const float* __restrict__ WihF, const float* __restrict__ WhhF,
          const float* __restrict__ biasF,
          const float* __restrict__ WihR, const float* __restrict__ WhhR,
          const float* __restrict__ biasR,
          float* __restrict__ hbF, float* __restrict__ hbR,   // each 2*32*512
          float* __restrict__ out,                            // (B,T,1024)
          unsigned* __restrict__ bar)                         // [2] arrive counters
{
    const int dir = blockIdx.x / NWG;
    const int wg  = blockIdx.x % NWG;
    const bool rev = (dir == 1);
    const float* Wih  = rev ? WihR : WihF;
    const float* Whh  = rev ? WhhR : WhhF;
    const float* bias = rev ? biasR : biasF;
    float* hb = rev ? hbR : hbF;
    unsigned* ctr = bar + dir;
    const int colOff = dir * H_;

    const int tid  = threadIdx.x;
    const int lane = tid & 31;          // wave32
    const int wave = tid >> 5;          // 0..15
    const int mt   = wave & 1;          // batch tile (0: rows 0-15, 1: rows 16-31)
    const int nt   = wave >> 1;         // 0..7 -> local gate cols [nt*16, nt*16+16)
    const int hi   = lane >> 4;         // lane half (selects K 0,1 vs 2,3)
    const int nn   = lane & 15;

    const int mrow  = mt*16 + nn;                         // batch row (A operand)
    const int gateT = nt >> 1;                            // 0..3 = i,f,g,o
    const int g     = gateT*H_ + wg*32 + (nt & 1)*16 + nn;// gate row for B column
    const float* wihRow = Wih + (size_t)g * (size_t)DX + 2*hi;
    const float* whhRow = Whh + (size_t)g * (size_t)H_  + 2*hi;

    __shared__ float P[B_ * 132];       // padded: 132 avoids bank conflicts

    // gate-stage mapping: 1024 (b,j) pairs over 512 threads x 2
    const int bA  = tid >> 5,          jlA = tid & 31;
    const int bB  = (tid + 512) >> 5,  jlB = tid & 31;
    const int jgA = wg*32 + jlA,       jgB = wg*32 + jlB;
    const float biA_i = bias[jgA], biA_f = bias[H_+jgA],
                biA_g = bias[2*H_+jgA], biA_o = bias[3*H_+jgA];
    const float biB_i = bias[jgB], biB_f = bias[H_+jgB],
                biB_g = bias[2*H_+jgB], biB_o = bias[3*H_+jgB];
    float cA = 0.f, cB = 0.f;

    const float* xBase = X + ((size_t)mrow * T_) * (size_t)DX + 2*hi;

    for (int s = 0; s < T_; ++s) {
        const int tx = rev ? (T_ - 1 - s) : s;
        const int rd = s & 1, wr = rd ^ 1;

        const float* xRow = xBase + (size_t)tx * DX;
        const float* hRow = hb + (size_t)rd*(B_*H_) + (size_t)mrow*H_ + 2*hi;

        v8f acc = {};
        #pragma unroll 8
        for (int k = 0; k < DX; k += 4) {
            v2f a = *(const v2f*)(xRow + k);
            v2f b = *(const v2f*)(wihRow + k);
            acc = __builtin_amdgcn_wmma_f32_16x16x4_f32(
                      false, a, false, b, (short)0, acc, false, false);
        }
        #pragma unroll 8
        for (int k = 0; k < H_; k += 4) {
            v2f a = *(const v2f*)(hRow + k);
            v2f b = *(const v2f*)(whhRow + k);
            acc = __builtin_amdgcn_wmma_f32_16x16x4_f32(
                      false, a, false, b, (short)0, acc, false, false);
        }

        // D layout: VGPR v holds rows v (lanes 0-15) / v+8 (lanes 16-31)
        #pragma unroll
        for (int v = 0; v < 8; ++v)
            P[(mt*16 + v + 8*hi)*132 + nt*16 + nn] = acc[v];
        __syncthreads();

        {   // pair A
            float pi = P[bA*132 +      jlA] + biA_i;
            float pf = P[bA*132 + 32 + jlA] + biA_f;
            float pg = P[bA*132 + 64 + jlA] + biA_g;
            float po = P[bA*132 + 96 + jlA] + biA_o;
            float ig = 1.f/(1.f + __expf(-pi));
            float fg = 1.f/(1.f + __expf(-pf));
            float gv = tanhf(pg);
            float og = 1.f/(1.f + __expf(-po));
            cA = fg*cA + ig*gv;
            float h = og * tanhf(cA);
            hb[(size_t)wr*(B_*H_) + (size_t)bA*H_ + jgA] = h;
            out[((size_t)bA*T_ + tx)*1024 + colOff + jgA] = h;
        }
        {   // pair B
            float pi = P[bB*132 +      jlB] + biB_i;
            float pf = P[bB*132 + 32 + jlB] + biB_f;
            float pg = P[bB*132 + 64 + jlB] + biB_g;
            float po = P[bB*132 + 96 + jlB] + biB_o;
            float ig = 1.f/(1.f + __expf(-pi));
            float fg = 1.f/(1.f + __expf(-pf));
            float gv = tanhf(pg);
            float og = 1.f/(1.f + __expf(-po));
            cB = fg*cB + ig*gv;
            float h = og * tanhf(cB);
            hb[(size_t)wr*(B_*H_) + (size_t)bB*H_ + jgB] = h;
            out[((size_t)bB*T_ + tx)*1024 + colOff + jgB] = h;
        }

        // device-scope per-direction barrier (monotonic arrive counter)
        __threadfence();
        __syncthreads();
        if (tid == 0) {
            atomicAdd(ctr, 1u);
            const unsigned need = (unsigned)(s + 1) * NWG;
            while (__hip_atomic_load(ctr, __ATOMIC_RELAXED,
                                     __HIP_MEMORY_SCOPE_AGENT) < need)
                __builtin_amdgcn_s_sleep(2);
        }
        __syncthreads();
        __threadfence();
    }
}

// ---------------- host launcher ----------------
extern "C" void kernel_launch(void* const* d_in, const int* in_sizes, int n_in,
                              void* d_out, int out_size, void* d_ws, size_t ws_size,
                              hipStream_t stream)
{
    const float* inputs   = (const float*)d_in[0];
    const float* emb      = (const float*)d_in[1];
    const float* w_ih_l0  = (const float*)d_in[2];
    const float* w_hh_l0  = (const float*)d_in[3];
    const float* b_ih_l0  = (const float*)d_in[4];
    const float* b_hh_l0  = (const float*)d_in[5];
    const float* w_ih_l0r = (const float*)d_in[6];
    const float* w_hh_l0r = (const float*)d_in[7];
    const float* b_ih_l0r = (const float*)d_in[8];
    const float* b_hh_l0r = (const float*)d_in[9];
    const float* w_ih_l1  = (const float*)d_in[10];
    const float* w_hh_l1  = (const float*)d_in[11];
    const float* b_ih_l1  = (const float*)d_in[12];
    const float* b_hh_l1  = (const float*)d_in[13];
    const float* w_ih_l1r = (const float*)d_in[14];
    const float* w_hh_l1r = (const float*)d_in[15];
    const float* b_ih_l1r = (const float*)d_in[16];
    const float* b_hh_l1r = (const float*)d_in[17];
    const int*   domain   = (const int*)d_in[18];

    float* ws = (float*)d_ws;
    float* e        = ws + OFF_E;
    unsigned* bar   = (unsigned*)(ws + OFF_BAR);
    float* hb_l0f   = ws + OFF_H;
    float* hb_l0r   = ws + OFF_H + 1ull*2*B_*H_;
    float* hb_l1f   = ws + OFF_H + 2ull*2*B_*H_;
    float* hb_l1r   = ws + OFF_H + 3ull*2*B_*H_;
    float* bias_l0f = ws + OFF_BIAS;
    float* bias_l0r = ws + OFF_BIAS + 1ull*G_;
    float* bias_l1f = ws + OFF_BIAS + 2ull*G_;
    float* bias_l1r = ws + OFF_BIAS + 3ull*G_;
    float* W0F_ih = ws + OFF_W0F_IH;  float* W0F_hh = ws + OFF_W0F_HH;
    float* W0B_ih = ws + OFF_W0B_IH;  float* W0B_hh = ws + OFF_W0B_HH;
    float* W1F_ih = ws + OFF_W1F_IH;  float* W1F_hh = ws + OFF_W1F_HH;
    float* W1B_ih = ws + OFF_W1B_IH;  float* W1B_hh = ws + OFF_W1B_HH;
    float* X1     = ws + OFF_X1;

    // zero e + barrier counters + h double-buffers (graph-capture safe)
    hipMemsetAsync(d_ws, 0, OFF_BIAS * sizeof(float), stream);

    embed_kernel<<<1, 32, 0, stream>>>(emb, domain, e);

    const int BL = 256;
    const int n_ih0 = G_*512, n_hh = G_*H_, n_ih1 = G_*1024;
    gen_weight<<<(n_ih0+BL-1)/BL, BL, 0, stream>>>(w_ih_l0,  e, W0F_ih, n_ih0);
    gen_weight<<<(n_hh +BL-1)/BL, BL, 0, stream>>>(w_hh_l0,  e, W0F_hh, n_hh);
    gen_weight<<<(n_ih0+BL-1)/BL, BL, 0, stream>>>(w_ih_l0r, e, W0B_ih, n_ih0);
    gen_weight<<<(n_hh +BL-1)/BL, BL, 0, stream>>>(w_hh_l0r, e, W0B_hh, n_hh);
    gen_weight<<<(n_ih1+BL-1)/BL, BL, 0, stream>>>(w_ih_l1,  e, W1F_ih, n_ih1);
    gen_weight<<<(n_hh +BL-1)/BL, BL, 0, stream>>>(w_hh_l1,  e, W1F_hh, n_hh);
    gen_weight<<<(n_ih1+BL-1)/BL, BL, 0, stream>>>(w_ih_l1r, e, W1B_ih, n_ih1);
    gen_weight<<<(n_hh +BL-1)/BL, BL, 0, stream>>>(w_hh_l1r, e, W1B_hh, n_hh);
    gen_bias<<<(G_+BL-1)/BL, BL, 0, stream>>>(b_ih_l0,  b_hh_l0,  e, bias_l0f, G_);
    gen_bias<<<(G_+BL-1)/BL, BL, 0, stream>>>(b_ih_l0r, b_hh_l0r, e, bias_l0r, G_);
    gen_bias<<<(G_+BL-1)/BL, BL, 0, stream>>>(b_ih_l1,  b_hh_l1,  e, bias_l1f, G_);
    gen_bias<<<(G_+BL-1)/BL, BL, 0, stream>>>(b_ih_l1r, b_hh_l1r, e, bias_l1r, G_);

    // layer 0: inputs (B,T,512) -> X1 (B,T,1024)
    lstm_scan<512><<<2*NWG, 512, 0, stream>>>(inputs,
                                              W0F_ih, W0F_hh, bias_l0f,
                                              W0B_ih, W0B_hh, bias_l0r,
                                              hb_l0f, hb_l0r, X1, bar + 0);
    // layer 1: X1 (B,T,1024) -> d_out (B,T,1024)
    lstm_scan<1024><<<2*NWG, 512, 0, stream>>>(X1,
                                               W1F_ih, W1F_hh, bias_l1f,
                                               W1B_ih, W1B_hh, bias_l1r,
                                               hb_l1f, hb_l1r, (float*)d_out, bar + 2);
}